// FRGCN_38439957299595
// MI455X (gfx1250) — compile-verified
//
#include <hip/hip_runtime.h>
#include <hip/hip_bf16.h>
#include <math.h>

#define NN   100000
#define IND  64
#define HID  32
#define FEAT 128
#define EE   1600000
#define GG   256

typedef __attribute__((ext_vector_type(2))) float v2f;
typedef __attribute__((ext_vector_type(4))) float v4f;
typedef __attribute__((ext_vector_type(8))) float v8f;

// ---------------------------------------------------------------------------
// Build Wcat[K x 128] = [W0 | W1 | W2 | loop],  W_r = sum_b comp[r,b]*basis[b]
// ---------------------------------------------------------------------------
__global__ __launch_bounds__(256) void prep_w_kernel(
    const float* __restrict__ basis,   // [2, K, 32]
    const float* __restrict__ comp,    // [3, 2]
    const float* __restrict__ loopw,   // [K, 32]
    float* __restrict__ Wcat,          // [K, 128]
    int K) {
  int t = blockIdx.x * 256 + threadIdx.x;
  if (t >= K * 128) return;
  int k   = t >> 7;
  int col = t & 127;
  int r   = col >> 5;
  int o   = col & 31;
  float v;
  if (r < 3) {
    v = comp[r * 2 + 0] * basis[(size_t)k * 32 + o]
      + comp[r * 2 + 1] * basis[(size_t)K * 32 + (size_t)k * 32 + o];
  } else {
    v = loopw[(size_t)k * 32 + o];
  }
  Wcat[t] = v;
}

// ---------------------------------------------------------------------------
// Tall-skinny GEMM: Out[N x 128] = A[N x K] (row stride LDA) @ Wcat[K x 128]
// using V_WMMA_F32_16X16X4_F32.  8 waves/block, each wave = 16 rows x 128 cols.
// Weights held COLUMN-major in LDS (padded stride K+2) so each B fragment is
// one aligned ds_load_b64 into an even VGPR pair (no repack movs).
// ---------------------------------------------------------------------------
template <int K, int LDA>
__global__ __launch_bounds__(256) void gemm_wmma_kernel(
    const float* __restrict__ A,
    const float* __restrict__ W,     // [K x 128] row-major in global
    float* __restrict__ Out,         // [N x 128]
    int nRows) {
  constexpr int KP = K + 2;          // padded column stride (floats)
  __shared__ float sWt[128 * KP];    // column-major: sWt[col*KP + k]

  int tid = threadIdx.x;
  for (int t = tid; t < K * 128; t += 256) {
    int k = t >> 7, c = t & 127;
    sWt[c * KP + k] = W[t];
  }
  __syncthreads();

  const int wave  = tid >> 5;
  const int lane  = tid & 31;
  const int row0  = (blockIdx.x * 8 + wave) * 16;
  const int m     = lane & 15;     // row within tile (A) / col within tile (B)
  const int khalf = lane >> 4;     // 0: K={0,1}, 1: K={2,3} within the 4-step

  // clamp load row so EXEC stays all-ones through the WMMAs (stores are guarded)
  int arow = row0 + m;
  if (arow >= nRows) arow = nRows - 1;
  const float* __restrict__ aptr = A + (size_t)arow * LDA;

  v8f acc[8];
#pragma unroll
  for (int t = 0; t < 8; ++t) acc[t] = (v8f){0.f, 0.f, 0.f, 0.f, 0.f, 0.f, 0.f, 0.f};

#pragma unroll
  for (int k0 = 0; k0 < K; k0 += 4) {
    const int ks = k0 + khalf * 2;
    v2f afrag = *(const v2f*)(aptr + ks);
#pragma unroll
    for (int ct = 0; ct < 8; ++ct) {
      // contiguous {W[ks][col], W[ks+1][col]} -> single aligned ds_load_b64
      v2f bfrag = *(const v2f*)(&sWt[(ct * 16 + m) * KP + ks]);
      acc[ct] = __builtin_amdgcn_wmma_f32_16x16x4_f32(
          false, afrag, false, bfrag, (short)0, acc[ct], false, false);
    }
  }

  // C/D layout: VGPR i -> row (khalf*8 + i), lanes%16 -> col within tile
  const int mtop = khalf * 8;
#pragma unroll
  for (int ct = 0; ct < 8; ++ct) {
#pragma unroll
    for (int i = 0; i < 8; ++i) {
      int r = row0 + mtop + i;
      if (r < nRows) Out[(size_t)r * 128 + ct * 16 + m] = acc[ct][i];
    }
  }
}

// ---------------------------------------------------------------------------
// agg[n][o] = tmp[n][96+o] + bias[o]   (self-loop slice + bias)
// ---------------------------------------------------------------------------
__global__ __launch_bounds__(256) void init_agg_kernel(
    const float* __restrict__ tmp, const float* __restrict__ bias,
    float* __restrict__ agg, int n) {
  int t = blockIdx.x * 256 + threadIdx.x;
  if (t >= n * HID) return;
  int node = t >> 5, o = t & 31;
  agg[t] = tmp[(size_t)node * 128 + 96 + o] + bias[o];
}

// ---------------------------------------------------------------------------
// edge scatter: agg[dst] += tmp[src][etype*32 .. +32]   (float4, 8 lanes/edge)
// NT loads: message stream is L2-resident; keep it out of L0.
// ---------------------------------------------------------------------------
__global__ __launch_bounds__(256) void edge_kernel(
    const float* __restrict__ tmp,
    const int* __restrict__ src, const int* __restrict__ dst,
    const int* __restrict__ etype, float* __restrict__ agg, int E) {
  int t = blockIdx.x * 256 + threadIdx.x;
  if (t >= E * 8) return;
  int e = t >> 3;
  int c = (t & 7) * 4;
  // run-ahead prefetch of the index streams (global_prefetch_b8)
  if (e + 8192 < E) {
    __builtin_prefetch(src + e + 8192, 0, 1);
    __builtin_prefetch(dst + e + 8192, 0, 1);
    __builtin_prefetch(etype + e + 8192, 0, 1);
  }
  int s = src[e], d = dst[e], r = etype[e];
  const v4f msg =
      __builtin_nontemporal_load((const v4f*)(tmp + (size_t)s * 128 + r * 32 + c));
  float* ap = agg + (size_t)d * 32 + c;
  atomicAdd(ap + 0, msg.x);
  atomicAdd(ap + 1, msg.y);
  atomicAdd(ap + 2, msg.z);
  atomicAdd(ap + 3, msg.w);
}

// ---------------------------------------------------------------------------
// feat[n][layer*32+o] = tanh(agg[n][o])  (next layer reads feat with lda=128)
// ---------------------------------------------------------------------------
__global__ __launch_bounds__(256) void act_kernel(
    const float* __restrict__ agg, float* __restrict__ feat, int layer, int n) {
  int t = blockIdx.x * 256 + threadIdx.x;
  if (t >= n * HID) return;
  int node = t >> 5, o = t & 31;
  feat[(size_t)node * 128 + layer * 32 + o] = tanhf(agg[t]);
}

// ---------------------------------------------------------------------------
// zero pooled + counts
// ---------------------------------------------------------------------------
__global__ __launch_bounds__(256) void zero_kernel(float* __restrict__ pooled,
                                                   float* __restrict__ counts) {
  int t = blockIdx.x * 256 + threadIdx.x;
  if (t < GG * 128) pooled[t] = 0.f;
  if (t < GG) counts[t] = 0.f;
}

// ---------------------------------------------------------------------------
// per-graph segment sum of feat + node counts
// ---------------------------------------------------------------------------
__global__ __launch_bounds__(256) void pool_kernel(
    const float* __restrict__ feat, const int* __restrict__ gid,
    float* __restrict__ pooled, float* __restrict__ counts, int n) {
  long long t = (long long)blockIdx.x * 256 + threadIdx.x;
  if (t >= (long long)n * 128) return;
  int node = (int)(t >> 7), c = (int)(t & 127);
  int g = gid[node];
  atomicAdd(pooled + (size_t)g * 128 + c, feat[t]);
  if (c == 0) atomicAdd(counts + g, 1.0f);
}

// ---------------------------------------------------------------------------
// final MLP: relu(pooled/cnt @ lin1 + b1) @ lin2 + b2 -> sigmoid
// ---------------------------------------------------------------------------
__global__ __launch_bounds__(64) void mlp_kernel(
    const float* __restrict__ pooled, const float* __restrict__ counts,
    const float* __restrict__ w1, const float* __restrict__ b1,
    const float* __restrict__ w2, const float* __restrict__ b2,
    float* __restrict__ out) {
  __shared__ float p[128];
  __shared__ float hdn[64];
  int g = blockIdx.x, j = threadIdx.x;
  float inv = 1.0f / fmaxf(counts[g], 1.0f);
  p[j]      = pooled[(size_t)g * 128 + j] * inv;
  p[j + 64] = pooled[(size_t)g * 128 + 64 + j] * inv;
  __syncthreads();
  float acc = b1[j];
  for (int c = 0; c < 128; ++c) acc += p[c] * w1[c * 64 + j];
  hdn[j] = fmaxf(acc, 0.0f);
  __syncthreads();
  if (j == 0) {
    float s = b2[0];
    for (int k = 0; k < 64; ++k) s += hdn[k] * w2[k];
    out[g] = 1.0f / (1.0f + expf(-s));
  }
}

// ---------------------------------------------------------------------------
extern "C" void kernel_launch(void* const* d_in, const int* in_sizes, int n_in,
                              void* d_out, int out_size, void* d_ws, size_t ws_size,
                              hipStream_t stream) {
  const float* x       = (const float*)d_in[0];   // [N,64]
  const float* basis1  = (const float*)d_in[1];   // [2,64,32]
  const float* basis_r = (const float*)d_in[2];   // [3,2,32,32]
  const float* w_comp  = (const float*)d_in[3];   // [4,3,2]
  const float* loop1   = (const float*)d_in[4];   // [64,32]
  const float* loop_r  = (const float*)d_in[5];   // [3,32,32]
  const float* bias    = (const float*)d_in[6];   // [4,32]
  const float* lin1_w  = (const float*)d_in[7];   // [128,64]
  const float* lin1_b  = (const float*)d_in[8];   // [64]
  const float* lin2_w  = (const float*)d_in[9];   // [64,1]
  const float* lin2_b  = (const float*)d_in[10];  // [1]
  const int*   src     = (const int*)d_in[11];    // [E]
  const int*   dst     = (const int*)d_in[12];    // [E]
  const int*   etype   = (const int*)d_in[13];    // [E]
  const int*   gid     = (const int*)d_in[14];    // [N]

  const int n = in_sizes[0] / IND;   // 100000
  const int E = in_sizes[11];        // 1600000

  // workspace partition (floats)
  float* ws     = (float*)d_ws;
  float* tmp    = ws;                           // N*128
  float* agg    = tmp + (size_t)n * 128;        // N*32
  float* feat   = agg + (size_t)n * 32;         // N*128
  float* Wcat   = feat + (size_t)n * 128;       // 64*128
  float* pooled = Wcat + 64 * 128;              // G*128
  float* counts = pooled + (size_t)GG * 128;    // G

  const int nvBlocks   = (n * HID + 255) / 256;
  const int edgeBlocks = (E * 8 + 255) / 256;
  const int gemmBlocks = (n + 127) / 128;       // 128 rows per block

  for (int L = 0; L < 4; ++L) {
    const float* basis = (L == 0) ? basis1 : basis_r + (size_t)(L - 1) * 2 * 32 * 32;
    const float* comp  = w_comp + L * 6;
    const float* loopw = (L == 0) ? loop1 : loop_r + (size_t)(L - 1) * 32 * 32;
    const int K = (L == 0) ? 64 : 32;

    prep_w_kernel<<<(K * 128 + 255) / 256, 256, 0, stream>>>(basis, comp, loopw, Wcat, K);

    if (L == 0) {
      gemm_wmma_kernel<64, 64><<<gemmBlocks, 256, 0, stream>>>(x, Wcat, tmp, n);
    } else {
      gemm_wmma_kernel<32, 128><<<gemmBlocks, 256, 0, stream>>>(
          feat + (L - 1) * 32, Wcat, tmp, n);
    }

    init_agg_kernel<<<nvBlocks, 256, 0, stream>>>(tmp, bias + L * 32, agg, n);
    edge_kernel<<<edgeBlocks, 256, 0, stream>>>(tmp, src, dst, etype, agg, E);
    act_kernel<<<nvBlocks, 256, 0, stream>>>(agg, feat, L, n);
  }

  zero_kernel<<<(GG * 128 + 255) / 256, 256, 0, stream>>>(pooled, counts);
  pool_kernel<<<(int)(((long long)n * 128 + 255) / 256), 256, 0, stream>>>(
      feat, gid, pooled, counts, n);
  mlp_kernel<<<GG, 64, 0, stream>>>(pooled, counts, lin1_w, lin1_b, lin2_w, lin2_b,
                                    (float*)d_out);
}